// HopfieldLayer_35699768165230
// MI455X (gfx1250) — compile-verified
//
#include <hip/hip_runtime.h>
#include <math.h>
#include <stdint.h>

// ---------------- problem constants ----------------
#define DIM 512
#define N_ELEM (DIM * DIM)

// solver constants (match reference)
constexpr float DT0         = 0.1f;
constexpr float TOL         = 0.01f;
constexpr float MIN_DT      = 0.1f;
constexpr float MAX_DELTA_T = 50.0f;
constexpr float VEL_TOL     = 0.05f;
#define MAX_STEPS 32

// RKF56 coefficients
constexpr float A21 = 0.25f;
constexpr float A31 = 3.0f/32.0f,  A32 = 9.0f/32.0f;
constexpr float A41 = 1932.0f/2197.0f, A42 = -7200.0f/2197.0f, A43 = 7296.0f/2197.0f;
constexpr float A51 = 439.0f/216.0f, A52 = -8.0f, A53 = 3680.0f/513.0f, A54 = -845.0f/4104.0f;
constexpr float A61 = -8.0f/27.0f, A62 = 2.0f, A63 = -3544.0f/2565.0f, A64 = 1859.0f/4104.0f, A65 = -11.0f/40.0f;
constexpr float B1c = 16.0f/135.0f, B3c = 6656.0f/12825.0f, B4c = 28561.0f/56430.0f, B5c = -9.0f/50.0f, B6c = 2.0f/55.0f;
constexpr float C1c = 25.0f/216.0f, C3c = 1408.0f/2565.0f, C4c = 2197.0f/4104.0f, C5c = -1.0f/5.0f;

// ---------------- vector types ----------------
typedef __attribute__((ext_vector_type(2))) float        v2f;
typedef __attribute__((ext_vector_type(8))) float        v8f;
typedef __attribute__((ext_vector_type(4))) unsigned int u32x4;
typedef __attribute__((ext_vector_type(8))) int          i32x8;
typedef __attribute__((ext_vector_type(4))) int          i32x4;

__device__ __forceinline__ v8f wmma_f32(v2f a, v2f b, v8f c) {
    // V_WMMA_F32_16X16X4_F32 : D = A(16x4) x B(4x16) + C(16x16), all fp32
    return __builtin_amdgcn_wmma_f32_16x16x4_f32(
        /*neg_a=*/false, a, /*neg_b=*/false, b,
        /*c_mod=*/(short)0, c, /*reuse_a=*/false, /*reuse_b=*/false);
}

// ---------------- Tensor Data Mover: 2D tile -> LDS ----------------
// Builds a D# per CDNA5 ISA ch.8 and issues TENSOR_LOAD_TO_LDS.
//   - data_size = 4 bytes (code 2)
//   - 2D tensor: tile_dim2 = 0, groups 2/3 zero
//   - optional LDS row padding: pad_interval code c -> pad after 2^(c+1) DWORDs,
//     pad_amount code 0 -> 1 DWORD
__device__ __forceinline__ void tdm_load_2d(unsigned lds_off,
                                            const float* gptr,
                                            unsigned tensor_d0, unsigned tensor_d1,
                                            unsigned d0_stride,
                                            unsigned tile_d0, unsigned tile_d1,
                                            unsigned pad_en, unsigned pad_iv)
{
    const unsigned long long ga = (unsigned long long)(uintptr_t)gptr;
    u32x4 g0;
    g0[0] = 1u;                                            // count=1, user descriptor
    g0[1] = lds_off;                                       // LDS byte address
    g0[2] = (unsigned)(ga & 0xFFFFFFFFu);                  // global_addr[31:0]
    g0[3] = (unsigned)((ga >> 32) & 0x1FFFFFFu)            // global_addr[56:32]
          | (2u << 30);                                    // type = 2 (image)
    i32x8 g1;
    g1[0] = (int)((2u << 16) | (pad_en << 20) | (pad_iv << 22)); // data_size=4B, pad_amount code 0
    g1[1] = (int)((tensor_d0 & 0xFFFFu) << 16);            // tensor_dim0[15:0] at bits [63:48]
    g1[2] = (int)(((tensor_d0 >> 16) & 0xFFFFu) | ((tensor_d1 & 0xFFFFu) << 16));
    g1[3] = (int)(((tensor_d1 >> 16) & 0xFFFFu) | ((tile_d0 & 0xFFFFu) << 16));
    g1[4] = (int)(tile_d1 & 0xFFFFu);                      // tile_dim1, tile_dim2=0
    g1[5] = (int)d0_stride;                                // tensor_dim0_stride[31:0]
    g1[6] = 0;                                             // stride[47:32]=0, dim1_stride lo=0
    g1[7] = 0;
    const i32x4 z4 = {0, 0, 0, 0};
    const i32x8 z8 = {0, 0, 0, 0, 0, 0, 0, 0};
    // 6-arg toolchain form: (g0, g1, group2, group3, extra group, cpol)
    __builtin_amdgcn_tensor_load_to_lds(g0, g1, z4, z4, z8, 0);
}

// ---------------- fused pvf GEMM (TDM -> LDS -> WMMA) ----------------
// MODE 0: out = 1 - tanh(A@W + bias)^2
// MODE 1: out = -(2*state + A@W^T) / DIM
// Block = 128 threads (4 waves); WG computes a 64x64 tile; wave owns 32x32
// (2x2 WMMA f32 accumulators). K is consumed in 32-wide chunks, double-buffered
// in LDS via the Tensor Data Mover (wave 0 issues, TENSORcnt synchronizes).
template <int MODE>
__global__ __launch_bounds__(128)
void pvf_gemm_kernel(const float* __restrict__ A,
                     const float* __restrict__ W,
                     const float* __restrict__ bias,   // MODE 0 only
                     const float* __restrict__ state,  // MODE 1 only
                     float* __restrict__ out)
{
    constexpr int KC = 32;            // K chunk
    constexpr int AS = 33;            // padded LDS row stride for 32-wide rows
    constexpr int BS0 = 65;           // padded LDS row stride for 64-wide rows (MODE 0)
    __shared__ float ldsA[2][64 * AS];        // A tile: [m][k], 64 rows
    __shared__ float ldsB[2][64 * AS];        // MODE0: [k][n] 32x65 (fits); MODE1: [n][k] 64x33

    const int lane = threadIdx.x & 31;
    const int wave = threadIdx.x >> 5;
    const int m0 = blockIdx.y * 64;
    const int n0 = blockIdx.x * 64;
    const int wm = (wave >> 1) * 32;          // wave row offset within WG tile
    const int wn = (wave & 1) * 32;           // wave col offset within WG tile
    const int l  = lane & 15;
    const int kh = (lane >> 4) << 1;          // 0 or 2

    auto issue = [&](int c) {
        const int buf = c & 1;
        // A tile: rows m0..m0+63, cols c*KC..c*KC+31, LDS stride 33 (pad every 32 DW)
        tdm_load_2d((unsigned)(uintptr_t)&ldsA[buf][0],
                    A + (size_t)m0 * DIM + c * KC,
                    DIM, DIM, DIM, KC, 64, 1, 4);
        if (MODE == 0) {
            // B = W (k-major): rows c*KC..+31, cols n0..n0+63, LDS stride 65 (pad every 64 DW)
            tdm_load_2d((unsigned)(uintptr_t)&ldsB[buf][0],
                        W + (size_t)(c * KC) * DIM + n0,
                        DIM, DIM, DIM, 64, KC, 1, 5);
        } else {
            // B = W^T: global layout is already [n][k] -> rows n0..n0+63, cols c*KC..+31
            tdm_load_2d((unsigned)(uintptr_t)&ldsB[buf][0],
                        W + (size_t)n0 * DIM + c * KC,
                        DIM, DIM, DIM, KC, 64, 1, 4);
        }
    };

    if (wave == 0) issue(0);

    v8f acc[2][2];
    acc[0][0] = v8f{}; acc[0][1] = v8f{}; acc[1][0] = v8f{}; acc[1][1] = v8f{};

    constexpr int NCHUNK = DIM / KC;
    for (int c = 0; c < NCHUNK; ++c) {
        __syncthreads();   // all waves done reading buffer (c+1)&1 from iteration c-1
        if (wave == 0) {
            if (c + 1 < NCHUNK) {
                issue(c + 1);                               // prefetch next chunk
                __builtin_amdgcn_s_wait_tensorcnt(2);       // chunk c's pair complete
            } else {
                __builtin_amdgcn_s_wait_tensorcnt(0);
            }
        }
        __syncthreads();   // chunk c tiles visible to all waves

        const float* As = &ldsA[c & 1][0];
        const float* Bs = &ldsB[c & 1][0];
        #pragma unroll
        for (int kk = 0; kk < KC; kk += 4) {
            v2f a0, a1, b0, b1;
            const int ar0 = (wm + l) * AS + kk + kh;
            const int ar1 = (wm + 16 + l) * AS + kk + kh;
            a0[0] = As[ar0];     a0[1] = As[ar0 + 1];
            a1[0] = As[ar1];     a1[1] = As[ar1 + 1];
            if (MODE == 0) {
                b0[0] = Bs[(kk + kh) * BS0 + wn + l];
                b0[1] = Bs[(kk + kh + 1) * BS0 + wn + l];
                b1[0] = Bs[(kk + kh) * BS0 + wn + 16 + l];
                b1[1] = Bs[(kk + kh + 1) * BS0 + wn + 16 + l];
            } else {
                const int br0 = (wn + l) * AS + kk + kh;
                const int br1 = (wn + 16 + l) * AS + kk + kh;
                b0[0] = Bs[br0]; b0[1] = Bs[br0 + 1];
                b1[0] = Bs[br1]; b1[1] = Bs[br1 + 1];
            }
            acc[0][0] = wmma_f32(a0, b0, acc[0][0]);
            acc[0][1] = wmma_f32(a0, b1, acc[0][1]);
            acc[1][0] = wmma_f32(a1, b0, acc[1][0]);
            acc[1][1] = wmma_f32(a1, b1, acc[1][1]);
        }
    }

    // C/D layout: VGPR j, lanes 0-15 -> (M=j, N=lane); lanes 16-31 -> (M=j+8)
    const int rbase = (lane >> 4) * 8;
    #pragma unroll
    for (int ti = 0; ti < 2; ++ti)
    #pragma unroll
    for (int tj = 0; tj < 2; ++tj)
    #pragma unroll
    for (int j = 0; j < 8; ++j) {
        const int row = m0 + wm + 16 * ti + rbase + j;
        const int col = n0 + wn + 16 * tj + l;
        float v = acc[ti][tj][j];
        if (MODE == 0) {
            float z  = v + bias[col];
            float th = tanhf(z);
            out[row * DIM + col] = 1.0f - th * th;
        } else {
            out[row * DIM + col] =
                -(2.0f * state[row * DIM + col] + v) * (1.0f / (float)DIM);
        }
    }
}

// ---------------- elementwise / reduction kernels ----------------
// Scalar slots in workspace: S[0]=t S[1]=dt S[2]=done S[3]=err S[4]=velsum S[5]=accept

__global__ void combine_kernel(float* __restrict__ out, const float* __restrict__ x,
                               const float* k1, const float* k2, const float* k3,
                               const float* k4, const float* k5,
                               float c1, float c2, float c3, float c4, float c5,
                               const float* __restrict__ S)
{
    const float dt = S[1];
    const int i = blockIdx.x * blockDim.x + threadIdx.x;
    if (i >= N_ELEM) return;
    float a = c1 * k1[i];
    if (k2) a += c2 * k2[i];
    if (k3) a += c3 * k3[i];
    if (k4) a += c4 * k4[i];
    if (k5) a += c5 * k5[i];
    out[i] = fmaf(dt, a, x[i]);
}

__device__ __forceinline__ float wave_max32(float v) {
    #pragma unroll
    for (int off = 16; off; off >>= 1)
        v = fmaxf(v, __shfl_xor(v, off, 32));
    return v;
}

__global__ void finalize_kernel(float* __restrict__ xh, const float* __restrict__ x,
                                const float* k1, const float* k3, const float* k4,
                                const float* k5, const float* k6, float* S)
{
    const float dt = S[1];
    const int i = blockIdx.x * blockDim.x + threadIdx.x;
    float lm = 0.0f;
    if (i < N_ELEM) {
        float shi = B1c * k1[i] + B3c * k3[i] + B4c * k4[i] + B5c * k5[i] + B6c * k6[i];
        float slo = C1c * k1[i] + C3c * k3[i] + C4c * k4[i] + C5c * k5[i];
        float hi = fmaf(dt, shi, x[i]);
        float lo = fmaf(dt, slo, x[i]);
        xh[i] = hi;
        lm = fabsf(hi - lo);
    }
    lm = wave_max32(lm);
    __shared__ float red[8];
    const int lane = threadIdx.x & 31, w = threadIdx.x >> 5;
    if (lane == 0) red[w] = lm;
    __syncthreads();
    if (threadIdx.x == 0) {
        float m = red[0];
        for (int q = 1; q < (int)(blockDim.x >> 5); ++q) m = fmaxf(m, red[q]);
        atomicMax((int*)&S[3], __float_as_int(m));  // m >= 0 -> int order == float order
    }
}

__global__ void select_kernel(float* __restrict__ x, const float* __restrict__ xh,
                              const float* __restrict__ S)
{
    if (S[2] != 0.0f || S[5] == 0.0f) return;   // frozen or rejected
    const int i = blockIdx.x * blockDim.x + threadIdx.x;
    if (i < N_ELEM) x[i] = xh[i];
}

// per-row max of |v|, mean accumulated into S[4] (one wave per row)
__global__ void rowmax_kernel(const float* __restrict__ v, float* S)
{
    const int lane = threadIdx.x & 31, w = threadIdx.x >> 5;
    const int row = blockIdx.x * (blockDim.x >> 5) + w;
    if (row >= DIM) return;
    const float* p = v + row * DIM;
    float m = 0.0f;
    for (int c = lane; c < DIM; c += 32) m = fmaxf(m, fabsf(p[c]));
    m = wave_max32(m);
    if (lane == 0) atomicAdd(&S[4], m);
}

__global__ void scalarA_kernel(float* S)   // accept / t / dt controller
{
    if (threadIdx.x != 0) return;
    const float t = S[0], dt = S[1], done = S[2], err = S[3];
    const bool accept = (err < TOL) || (dt <= MIN_DT);
    float factor = 0.9f * powf(TOL / (err + 1e-12f), 0.2f);
    factor = fminf(fmaxf(factor, 0.2f), 2.0f);
    const float dtn = fmaxf(dt * factor, MIN_DT);
    if (done == 0.0f) {
        if (accept) S[0] = t + dt;
        S[1] = dtn;
    }
    S[5] = accept ? 1.0f : 0.0f;
}

__global__ void scalarB_kernel(float* S)   // stop condition
{
    if (threadIdx.x != 0) return;
    if (S[2] == 0.0f) {
        const float vel = S[4] * (1.0f / (float)DIM);
        if (S[0] > MAX_DELTA_T || vel < VEL_TOL) S[2] = 1.0f;
    }
}

__global__ void init_copy_kernel(float* __restrict__ x, const float* __restrict__ x0)
{
    const int i = blockIdx.x * blockDim.x + threadIdx.x;
    if (i < N_ELEM) x[i] = x0[i];
}

__global__ void init_scalars_kernel(float* S)
{
    if (threadIdx.x != 0) return;
    S[0] = 0.0f; S[1] = DT0; S[2] = 0.0f; S[3] = 0.0f; S[4] = 0.0f; S[5] = 0.0f;
}

__global__ void prestep_kernel(float* S)
{
    if (threadIdx.x != 0) return;
    ((int*)S)[3] = 0;   // err  (stored as non-negative float bits)
    S[4] = 0.0f;        // velsum
}

// ---------------- host orchestration ----------------
extern "C" void kernel_launch(void* const* d_in, const int* in_sizes, int n_in,
                              void* d_out, int out_size, void* d_ws, size_t ws_size,
                              hipStream_t stream)
{
    const float* x0 = (const float*)d_in[0];
    const float* W  = (const float*)d_in[1];
    const float* bv = (const float*)d_in[2];
    float* X  = (float*)d_out;          // current state lives in d_out
    float* ws = (float*)d_ws;

    float* K1    = ws + 0 * (size_t)N_ELEM;
    float* K2    = ws + 1 * (size_t)N_ELEM;
    float* K3    = ws + 2 * (size_t)N_ELEM;
    float* K4    = ws + 3 * (size_t)N_ELEM;
    float* K5    = ws + 4 * (size_t)N_ELEM;
    float* K6    = ws + 5 * (size_t)N_ELEM;
    float* SECH2 = ws + 6 * (size_t)N_ELEM;
    float* XS    = ws + 7 * (size_t)N_ELEM;
    float* XH    = ws + 8 * (size_t)N_ELEM;
    float* S     = ws + 9 * (size_t)N_ELEM;   // 16 scalar slots

    const dim3 ggrid(8, 8);
    const int  gblk = 128;
    const int  eblk = 256;
    const int  egrid = (N_ELEM + eblk - 1) / eblk;

    auto pvf = [&](const float* st, float* kout) {
        pvf_gemm_kernel<0><<<ggrid, gblk, 0, stream>>>(st, W, bv, nullptr, SECH2);
        pvf_gemm_kernel<1><<<ggrid, gblk, 0, stream>>>(SECH2, W, nullptr, st, kout);
    };

    // init: X = x0 ; scalars ; K1 = pvf(X)
    init_copy_kernel<<<egrid, eblk, 0, stream>>>(X, x0);
    init_scalars_kernel<<<1, 32, 0, stream>>>(S);
    pvf(X, K1);

    for (int step = 0; step < MAX_STEPS; ++step) {
        prestep_kernel<<<1, 32, 0, stream>>>(S);

        combine_kernel<<<egrid, eblk, 0, stream>>>(XS, X, K1, nullptr, nullptr, nullptr, nullptr,
                                                   A21, 0.f, 0.f, 0.f, 0.f, S);
        pvf(XS, K2);
        combine_kernel<<<egrid, eblk, 0, stream>>>(XS, X, K1, K2, nullptr, nullptr, nullptr,
                                                   A31, A32, 0.f, 0.f, 0.f, S);
        pvf(XS, K3);
        combine_kernel<<<egrid, eblk, 0, stream>>>(XS, X, K1, K2, K3, nullptr, nullptr,
                                                   A41, A42, A43, 0.f, 0.f, S);
        pvf(XS, K4);
        combine_kernel<<<egrid, eblk, 0, stream>>>(XS, X, K1, K2, K3, K4, nullptr,
                                                   A51, A52, A53, A54, 0.f, S);
        pvf(XS, K5);
        combine_kernel<<<egrid, eblk, 0, stream>>>(XS, X, K1, K2, K3, K4, K5,
                                                   A61, A62, A63, A64, A65, S);
        pvf(XS, K6);

        finalize_kernel<<<egrid, eblk, 0, stream>>>(XH, X, K1, K3, K4, K5, K6, S);
        scalarA_kernel<<<1, 32, 0, stream>>>(S);
        select_kernel<<<egrid, eblk, 0, stream>>>(X, XH, S);

        // stop-condition pvf doubles as next step's k1
        pvf(X, K1);
        rowmax_kernel<<<DIM / 8, 256, 0, stream>>>(K1, S);
        scalarB_kernel<<<1, 32, 0, stream>>>(S);
    }
}